// GCNConv_57767310131235
// MI455X (gfx1250) — compile-verified
//
#include <hip/hip_runtime.h>

typedef float v2f __attribute__((ext_vector_type(2)));
typedef float v8f __attribute__((ext_vector_type(8)));

#define DIN   128
#define DOUT  32
#define LDS_K (DIN + 4)   // padded stride: bank = (4*col + k) % 64 -> conflict-free b64 reads

// -------- Kernel 1: HW[N,32] = H[N,128] @ W[128,32] via V_WMMA_F32_16X16X4_F32
// Block = 256 threads = 8 waves. Each wave computes a 16x32 output tile
// (two 16x16 f32 accumulators), looping K over 128 in steps of 4.
// W is staged TRANSPOSED in LDS so each B fragment is one contiguous ds_load_b64.
__global__ void __launch_bounds__(256) gcn_gemm(const float* __restrict__ H,
                                                const float* __restrict__ W,
                                                float* __restrict__ HW,
                                                int n) {
    __shared__ float sWt[DOUT * LDS_K];  // ~16.5 KB, transposed: sWt[c*LDS_K + k] = W[k][c]
    const int tid = threadIdx.x;
    for (int i = tid; i < DIN * DOUT; i += 256) {
        const int k = i >> 5;        // 0..127
        const int c = i & 31;        // 0..31
        sWt[c * LDS_K + k] = W[i];   // W is row-major [k][c]
    }
    __syncthreads();

    const int wave    = tid >> 5;
    const int lane    = tid & 31;
    const int rowBase = (blockIdx.x * 8 + wave) * 16;
    if (rowBase >= n) return;  // whole-wave exit keeps EXEC uniform for WMMA

    const int lr    = lane & 15;        // A: row-in-tile; B/C: column-in-tile
    const int khalf = (lane >> 4) * 2;  // K sub-offset: lanes 0-15 -> K{0,1}, 16-31 -> K{2,3}

    int arow = rowBase + lr;
    if (arow >= n) arow = n - 1;        // clamp for tail tiles
    const float* __restrict__ hrow = H   + (size_t)arow * DIN + khalf;
    const float* __restrict__ w0   = sWt + lr * LDS_K + khalf;         // col tile 0
    const float* __restrict__ w1   = sWt + (lr + 16) * LDS_K + khalf;  // col tile 1

    v8f acc0 = {};
    v8f acc1 = {};
#pragma unroll 4
    for (int k = 0; k < DIN; k += 4) {
        const v2f a  = *(const v2f*)(hrow + k);  // global b64
        const v2f b0 = *(const v2f*)(w0 + k);    // ds_load_b64, conflict-free
        const v2f b1 = *(const v2f*)(w1 + k);
        acc0 = __builtin_amdgcn_wmma_f32_16x16x4_f32(false, a, false, b0,
                                                     (short)0, acc0, false, false);
        acc1 = __builtin_amdgcn_wmma_f32_16x16x4_f32(false, a, false, b1,
                                                     (short)0, acc1, false, false);
    }

    // C/D layout: lanes 0-15 hold M=0..7 (VGPR i -> M=i), lanes 16-31 hold M=8..15
    const int mo = (lane >> 4) * 8;
    float* __restrict__ outBase = HW + (size_t)(rowBase + mo) * DOUT + lr;
    if (rowBase + 16 <= n) {
        // fast path: full tile, no per-element guards (N % 16 == 0 in practice)
#pragma unroll
        for (int i = 0; i < 8; ++i) {
            outBase[(size_t)i * DOUT]      = acc0[i];
            outBase[(size_t)i * DOUT + 16] = acc1[i];
        }
    } else {
#pragma unroll
        for (int i = 0; i < 8; ++i) {
            if (rowBase + mo + i < n) {
                outBase[(size_t)i * DOUT]      = acc0[i];
                outBase[(size_t)i * DOUT + 16] = acc1[i];
            }
        }
    }
}

// -------- Kernel 2: out = A_csr @ HW. One wave per row; lane = output column.
// Edge metadata (colIdx/val) addresses are wave-uniform -> scalar (SMEM) loads.
// Each neighbor gather is one coalesced 128B row of HW (resident in 192MB L2).
__global__ void __launch_bounds__(256) gcn_spmm(const float* __restrict__ HW,
                                                const int* __restrict__ rowPtr,
                                                const int* __restrict__ colIdx,
                                                const float* __restrict__ val,
                                                float* __restrict__ out,
                                                int n) {
    const int wv   = __builtin_amdgcn_readfirstlane(threadIdx.x >> 5);  // uniform wave id
    const int lane = threadIdx.x & 31;
    const int row  = blockIdx.x * 8 + wv;                               // wave-uniform
    if (row >= n) return;

    const int s = __builtin_amdgcn_readfirstlane(rowPtr[row]);
    const int e = __builtin_amdgcn_readfirstlane(rowPtr[row + 1]);

    float acc = 0.0f;
#pragma unroll 4
    for (int i = s; i < e; ++i) {
        const int   c = colIdx[i];   // uniform -> s_load
        const float v = val[i];      // uniform -> s_load
        acc += v * HW[(size_t)c * DOUT + lane];  // coalesced 128B gather
    }
    // written once, never re-read: nontemporal to keep L2 for the HW table
    __builtin_nontemporal_store(acc, out + (size_t)row * DOUT + lane);
}

extern "C" void kernel_launch(void* const* d_in, const int* in_sizes, int n_in,
                              void* d_out, int out_size, void* d_ws, size_t ws_size,
                              hipStream_t stream) {
    const float* H      = (const float*)d_in[0];  // [N, 128]
    const float* W      = (const float*)d_in[1];  // [128, 32]
    const int*   rowPtr = (const int*)d_in[2];    // [N+1]
    const int*   colIdx = (const int*)d_in[3];    // [E]
    const float* val    = (const float*)d_in[4];  // [E]
    float*       out    = (float*)d_out;          // [N, 32]
    float*       HW     = (float*)d_ws;           // scratch: N*32 floats (12.8 MB)

    const int N = in_sizes[2] - 1;

    const int rowTiles = (N + 15) / 16;           // one 16-row tile per wave
    const int blocks1  = (rowTiles + 7) / 8;      // 8 waves per block
    gcn_gemm<<<blocks1, 256, 0, stream>>>(H, W, HW, N);

    const int blocks2 = (N + 7) / 8;              // 8 rows (waves) per block
    gcn_spmm<<<blocks2, 256, 0, stream>>>(HW, rowPtr, colIdx, val, out, N);
}